// ResNetBasicHead_16836271800853
// MI455X (gfx1250) — compile-verified
//
#include <hip/hip_runtime.h>

// Problem constants (from reference)
#define B_DIM   64
#define C_IN    2304
#define THW     392            // 8*7*7, contiguous innermost per (b,c)
#define F_NUM   16
#define ROWS    (B_DIM * C_IN) // 147456 pooling rows
#define VEC_PER_ROW (THW / 4)  // 98 float4 per row (1568 B, 16B aligned)

typedef __attribute__((ext_vector_type(4))) float v4f;
typedef __attribute__((ext_vector_type(2))) float v2f;
typedef __attribute__((ext_vector_type(8))) float v8f;

// ---------------------------------------------------------------------------
// Stage 1: global average pool over THW. One wave32 per (b,c) row.
// 231 MB streamed once -> bandwidth bound (~10 us at 23.3 TB/s).
// ---------------------------------------------------------------------------
__global__ __launch_bounds__(256) void pool_kernel(const float* __restrict__ x,
                                                   float* __restrict__ pooled) {
    const int lane = threadIdx.x & 31;
    const int wave = threadIdx.x >> 5;                 // 8 waves per block
    const int row  = blockIdx.x * 8 + wave;            // one row per wave
    if (row >= ROWS) return;

    const v4f* __restrict__ p = (const v4f*)(x + (size_t)row * THW);

    float s = 0.0f;
    #pragma unroll 4
    for (int i = lane; i < VEC_PER_ROW; i += 32) {
        v4f v = p[i];                                  // global_load_b128
        s += v.x + v.y + v.z + v.w;
    }

    // wave32 butterfly reduction
    #pragma unroll
    for (int off = 16; off > 0; off >>= 1)
        s += __shfl_xor(s, off, 32);

    if (lane == 0)
        pooled[row] = s * (1.0f / (float)THW);
}

// ---------------------------------------------------------------------------
// Stage 2: out[b,f] = sum_c pooled[b,c] * W[f,c] + bias[f]
// Native fp32 WMMA: V_WMMA_F32_16X16X4_F32. One block, 4 waves; wave w
// handles batch rows [16w, 16w+16). K loop: 2304/4 = 576 WMMAs per wave.
//
// A (16x4 f32, 2 VGPRs): lanes 0-15 -> row M=lane, K=k0,k0+1;
//                        lanes 16-31 -> row M=lane-16, K=k0+2,k0+3.
// B (4x16 f32, 2 VGPRs): lanes 0-15 -> col N=lane, K=k0,k0+1;
//                        lanes 16-31 -> col N=lane-16, K=k0+2,k0+3.
// D (16x16 f32, 8 VGPRs): VGPR j -> M = j + 8*(lane>=16), N = lane&15.
// ---------------------------------------------------------------------------
__global__ __launch_bounds__(128) void head_gemm_kernel(const float* __restrict__ pooled,
                                                        const float* __restrict__ W,
                                                        const float* __restrict__ bias,
                                                        float* __restrict__ out) {
    const int lane  = threadIdx.x & 31;
    const int wave  = threadIdx.x >> 5;   // 0..3 -> batch tile
    const int m0    = wave * 16;
    const int n     = lane & 15;          // frame / B-column (also A row id)
    const int khalf = (lane >> 4) * 2;    // 0 for lanes 0-15, 2 for 16-31

    const float* __restrict__ arow = pooled + (size_t)(m0 + n) * C_IN + khalf;
    const float* __restrict__ brow = W      + (size_t)n         * C_IN + khalf;

    v8f acc = {};
    for (int k = 0; k < C_IN; k += 4) {
        v2f a, b;
        a.x = arow[k];
        a.y = arow[k + 1];
        b.x = brow[k];
        b.y = brow[k + 1];
        // 8 args: (neg_a, A, neg_b, B, c_mod, C, reuse_a, reuse_b)
        acc = __builtin_amdgcn_wmma_f32_16x16x4_f32(
            false, a, false, b, (short)0, acc, false, false);
    }

    const float bv = bias[n];
    const int mbase = (lane >> 4) * 8;    // lanes 16-31 hold M=j+8
    #pragma unroll
    for (int j = 0; j < 8; ++j)
        out[(size_t)(m0 + mbase + j) * F_NUM + n] = acc[j] + bv;
}

// ---------------------------------------------------------------------------
extern "C" void kernel_launch(void* const* d_in, const int* in_sizes, int n_in,
                              void* d_out, int out_size, void* d_ws, size_t ws_size,
                              hipStream_t stream) {
    const float* x  = (const float*)d_in[0];   // [64, 2304, 8, 7, 7]
    const float* W  = (const float*)d_in[1];   // [16, 2304]
    const float* bi = (const float*)d_in[2];   // [16]
    float* out      = (float*)d_out;           // [64, 16]
    float* pooled   = (float*)d_ws;            // [64, 2304] scratch (590 KB)

    (void)in_sizes; (void)n_in; (void)out_size; (void)ws_size;

    // 147456 rows / 8 waves per block
    pool_kernel<<<ROWS / 8, 256, 0, stream>>>(x, pooled);
    head_gemm_kernel<<<1, 128, 0, stream>>>(pooled, W, bi, out);
}